// TimeNormalization_61340722921852
// MI455X (gfx1250) — compile-verified
//
#include <hip/hip_runtime.h>

// Problem constants (match reference)
#define ALPHA_F 0.1f
#define DECAY_F 0.9f
#define EPS_F   1e-3f
#define B_DIM   32
#define T_DIM   8192
#define F_DIM   128

// Chunked-scan parameters: each chunk of CH timesteps is scanned independently
// after a WARM-step warm-up from zero state (0.9^512 ~ 4e-24, below fp32 eps).
// CH=1024 -> 8 chunks -> 256 blocks * 8 waves = 2048 waves for latency hiding.
#define CH_LEN   1024
#define WARM_LEN 512
#define NCHUNK   (T_DIM / CH_LEN)   // 8
#define PF_DIST  96                 // prefetch distance in timesteps (6 tiles)
#define TILE_F   (16 * F_DIM)       // floats per 16-timestep stride

#if __has_builtin(__builtin_amdgcn_rsqf)
#define FAST_RSQRT(v) __builtin_amdgcn_rsqf(v)   // raw v_rsq_f32; input >= EPS, no denorm guard needed
#else
#define FAST_RSQRT(v) rsqrtf(v)
#endif

typedef float v2f __attribute__((ext_vector_type(2)));
typedef float v8f __attribute__((ext_vector_type(8)));

__device__ __forceinline__ float powi09(int d) {
    // exact 0.9^d for small d >= 0, else 0 (lower-triangular mask incl. diagonal)
    if (d < 0) return 0.0f;
    float r = 1.0f;
    for (int i = 0; i < d; ++i) r *= DECAY_F;
    return r;
}

__global__ __launch_bounds__(256)
void TimeNorm_wmma_scan(const float* __restrict__ x,
                        const float* __restrict__ state0,
                        float* __restrict__ out) {
    const int lane  = threadIdx.x & 31;
    const int wave  = threadIdx.x >> 5;       // 8 waves/block, one 16-feature group each
    const int b     = blockIdx.x / NCHUNK;
    const int chunk = blockIdx.x % NCHUNK;
    const int fb    = wave * 16;              // feature group base
    const int n     = lane & 15;              // N column (and M row for A) per ISA layout
    const int hi    = lane >> 4;              // 0: lanes 0-15, 1: lanes 16-31
    const int h     = hi * 2;                 // K offset for A/B layouts

    // A = L (16x16 lower-triangular decay), split into 4 K-chunks of 16x4.
    // A layout (32-bit 16x4): lanes 0-15/16-31 both M=0..15; VGPR0 = K0(K2), VGPR1 = K1(K3).
    v2f A[4];
#pragma unroll
    for (int q = 0; q < 4; ++q) {
        A[q].x = powi09(n - (4 * q + h));
        A[q].y = powi09(n - (4 * q + h + 1));
    }
    // Carry WMMA: C[m,n] = c^(m+1)*carry[n] = Acar x Bcar (rank-1, K slot 0 only).
    // Acar[m,0] = c^(m+1): lanes 0-15 hold K=0 in .x; lanes 16-31 hold K=2/3 -> zero.
    v2f Acar = { hi ? 0.0f : powi09(n + 1), 0.0f };

    const float* xb = x + (size_t)b * T_DIM * F_DIM;
    float* st_out = out + (size_t)B_DIM * T_DIM * F_DIM;

    float carry_m, carry_s;
    int tstart;
    if (chunk == 0) {
        carry_m = state0[(size_t)b * F_DIM + fb + n];
        carry_s = state0[(size_t)B_DIM * F_DIM + (size_t)b * F_DIM + fb + n];
        tstart  = 0;
    } else {
        carry_m = 0.0f;   // warm-up from zero; error < 0.9^WARM ~ 4e-24
        carry_s = 0.0f;
        tstart  = chunk * CH_LEN;
        tstart -= WARM_LEN;
    }
    const int tmain = chunk * CH_LEN;
    const int tend  = tmain + CH_LEN;

    // Per-lane stream pointers, advanced by one tile (8 KB) per step so all tile
    // accesses use compile-time immediate offsets (fit 24-bit IOFFSET).
    const float* pBn = xb + (size_t)tstart * F_DIM + (size_t)h * F_DIM + fb + n;        // next B tile
    const float* pDc = xb + (size_t)tstart * F_DIM + (size_t)(8 * hi) * F_DIM + fb + n; // current D tile
    float*       poc = out + (size_t)b * T_DIM * F_DIM
                           + (size_t)tstart * F_DIM + (size_t)(8 * hi) * F_DIM + fb + n;
    const float* pfc = xb + (size_t)(tstart + PF_DIST) * F_DIM + (size_t)lane * F_DIM + fb;

    // Preload first tile.
    float bufA[8], bufB[8];
#pragma unroll
    for (int q = 0; q < 4; ++q) {
        bufA[2 * q]     = pBn[(4 * q)     * F_DIM];
        bufA[2 * q + 1] = pBn[(4 * q + 1) * F_DIM];
    }
    pBn += TILE_F;

    // One 16-timestep step: consume `cur`, preload `nxt`, all via immediate offsets.
    auto step = [&](int t0, float (&cur)[8], float (&nxt)[8]) {
        // L2 prefetch ~6 tiles ahead: 32 lanes cover rows t0+PF .. t0+PF+31
        if (t0 + PF_DIST <= T_DIM - 32) __builtin_prefetch(pfc, 0, 0);
        pfc += TILE_F;

        // double-buffer: issue next tile's loads before consuming current tile
        if (t0 + 16 < tend) {
#pragma unroll
            for (int q = 0; q < 4; ++q) {
                nxt[2 * q]     = pBn[(4 * q)     * F_DIM];
                nxt[2 * q + 1] = pBn[(4 * q + 1) * F_DIM];
            }
        }
        pBn += TILE_F;

        // Seed accumulators with the carry term via rank-1 WMMA (inline-0 C).
        v2f bcm = { hi ? 0.0f : carry_m, 0.0f };
        v2f bcs = { hi ? 0.0f : carry_s, 0.0f };
        v8f zero = {};
        v8f dm = __builtin_amdgcn_wmma_f32_16x16x4_f32(false, Acar, false, bcm, (short)0, zero, false, false);
        v8f ds = __builtin_amdgcn_wmma_f32_16x16x4_f32(false, Acar, false, bcs, (short)0, zero, false, false);

        // S = L*P + C via 4 chained K=4 WMMAs (x2 for mean / second moment)
#pragma unroll
        for (int q = 0; q < 4; ++q) {
            float pm0 = ALPHA_F * cur[2 * q];
            float pm1 = ALPHA_F * cur[2 * q + 1];
            v2f bm = {pm0, pm1};
            v2f bs = {pm0 * cur[2 * q], pm1 * cur[2 * q + 1]};
            dm = __builtin_amdgcn_wmma_f32_16x16x4_f32(false, A[q], false, bm, (short)0, dm, false, false);
            ds = __builtin_amdgcn_wmma_f32_16x16x4_f32(false, A[q], false, bs, (short)0, ds, false, false);
        }

        // main region only: normalize + NT-store (D-layout re-read hits WGP$)
        if (t0 >= tmain) {
#pragma unroll
            for (int v = 0; v < 8; ++v) {
                float xv   = pDc[v * F_DIM];
                float mean = dm[v];
                float var  = ds[v] - mean * mean;
                float res  = (xv - mean) * FAST_RSQRT(var + EPS_F);
                __builtin_nontemporal_store(res, &poc[v * F_DIM]);
            }
        }
        pDc += TILE_F;
        poc += TILE_F;

        // new carry = row M=15 of D (VGPR7, lanes 16-31); broadcast per column
        carry_m = __shfl(dm[7], 16 | n, 32);
        carry_s = __shfl(ds[7], 16 | n, 32);
    };

    // Iteration count (tend - tstart)/16 is 64 or 96 -> always even: ping-pong 2x.
    for (int t0 = tstart; t0 < tend; t0 += 32) {
        step(t0,      bufA, bufB);
        step(t0 + 16, bufB, bufA);
    }

    // ---------- final state (last chunk owns t = T-1) ----------
    if (chunk == NCHUNK - 1 && lane < 16) {
        st_out[(size_t)b * F_DIM + fb + n]                         = carry_m;
        st_out[(size_t)B_DIM * F_DIM + (size_t)b * F_DIM + fb + n] = carry_s;
    }
}

extern "C" void kernel_launch(void* const* d_in, const int* in_sizes, int n_in,
                              void* d_out, int out_size, void* d_ws, size_t ws_size,
                              hipStream_t stream) {
    (void)in_sizes; (void)n_in; (void)out_size; (void)d_ws; (void)ws_size;
    const float* x  = (const float*)d_in[0];
    const float* st = (const float*)d_in[1];
    float* out      = (float*)d_out;
    // 32 b * 8 chunks = 256 blocks; 256 threads = 8 waves, one 16-feature group each.
    TimeNorm_wmma_scan<<<dim3(B_DIM * NCHUNK), dim3(256), 0, stream>>>(x, st, out);
}